// SamplingGAT_40896678592677
// MI455X (gfx1250) — compile-verified
//
#include <hip/hip_runtime.h>
#include <math.h>

// ---------------- model dims ----------------
constexpr int N1 = 80000, N2 = 16000, N3 = 4096;
constexpr int E1 = 256000, E2 = 65536;

typedef float v2f __attribute__((ext_vector_type(2)));
typedef float v8f __attribute__((ext_vector_type(8)));

#if __has_builtin(__builtin_amdgcn_global_load_async_to_lds_b32)
#define HAS_ASYNC_LDS 1
typedef __attribute__((address_space(1))) int g_i32;  // global (AS1)
typedef __attribute__((address_space(3))) int l_i32;  // LDS (AS3)
#else
#define HAS_ASYNC_LDS 0
#endif

static __device__ __forceinline__ void waitAsync0() {
#if HAS_ASYNC_LDS
#if __has_builtin(__builtin_amdgcn_s_wait_asynccnt)
  __builtin_amdgcn_s_wait_asynccnt(0);
#else
  asm volatile("s_wait_asynccnt 0x0" ::: "memory");
#endif
#endif
}

static __device__ __forceinline__ void atomAddF(float* p, float v) {
  unsafeAtomicAdd(p, v);  // lowers to global_atomic_add_f32
}

// float atomic-max via integer atomics (IEEE ordering trick); amax initialized
// to -1e30f so both branches are safe.
static __device__ __forceinline__ void atomMaxF(float* addr, float val) {
  if (val >= 0.0f) atomicMax((int*)addr, __float_as_int(val));
  else             atomicMin((unsigned int*)addr, __float_as_uint(val));
}

static __device__ __forceinline__ v8f wmma4(v2f a, v2f b, v8f c) {
  return __builtin_amdgcn_wmma_f32_16x16x4_f32(false, a, false, b, (short)0, c, false, false);
}

// ---------------- WMMA f32 GEMM: C[M,N] = A[M,K] @ B[K,N], row-major ----------------
// Block = 256 threads = 8 waves; each wave computes a 32x64 tile of C (two 16x16
// M-subtiles x four N-subtiles), so a block covers 256 rows x 64 cols.
// B is staged through LDS in 128-deep K-chunks (32 KB), pre-swizzled into
// (K-pair, column) order so each lane's B fragment pair is one ds_load_2addr_b64.
// Staging uses GLOBAL_LOAD_ASYNC_TO_LDS_B32 when available (ASYNCcnt path),
// scattering directly into the swizzled layout via per-lane LDS addresses.
// Requires M%32==0, N%64==0, K%128==0 (true for all call sites).
__global__ void __launch_bounds__(256)
wmma_gemm_f32(const float* __restrict__ A, const float* __restrict__ B,
              float* __restrict__ C, int M, int N, int K) {
  constexpr int KC = 128;                  // K-chunk staged in LDS
  __shared__ float lsB[KC * 64];           // KC/2 x 64 pairs = 32 KB
  v2f* lsB2 = (v2f*)lsB;

  const int lane = threadIdx.x & 31;
  const int wave = threadIdx.x >> 5;
  const int grp  = lane >> 4;   // half-wave: selects K pair (A/B) and M rows 8..15 (C)
  const int l    = lane & 15;
  const int m0   = (blockIdx.x * 8 + wave) * 32;
  const int n0   = blockIdx.y * 64;
  const bool active = (m0 < M);            // wave-uniform

  v8f acc[2][4];
#pragma unroll
  for (int t = 0; t < 2; ++t)
#pragma unroll
    for (int j = 0; j < 4; ++j) acc[t][j] = (v8f){};

  const float* Arow0 = A + (size_t)(m0 + l) * K + 2 * grp;       // M-subtile 0
  const float* Arow1 = Arow0 + (size_t)16 * K;                   // M-subtile 1

  for (int kc = 0; kc < K; kc += KC) {
    if (active && kc + KC < K) {           // pull next A chunk toward the caches
      __builtin_prefetch(Arow0 + kc + KC, 0, 0);
      __builtin_prefetch(Arow1 + kc + KC, 0, 0);
    }
    __syncthreads();           // previous chunk's readers done
    // ---- cooperative stage of B[kc..kc+KC) x [n0..n0+64) as K-pairs ----
#pragma unroll
    for (int p = threadIdx.x; p < (KC / 2) * 64; p += 256) {
      int kp = p >> 6;                       // pair index within chunk
      int n  = p & 63;
      const float* Bp = B + (size_t)(kc + 2 * kp) * N + n0 + n;
#if HAS_ASYNC_LDS
      float* Lp = &lsB[2 * p];
      __builtin_amdgcn_global_load_async_to_lds_b32((g_i32*)Bp,       (l_i32*)Lp,       0, 0);
      __builtin_amdgcn_global_load_async_to_lds_b32((g_i32*)(Bp + N), (l_i32*)(Lp + 1), 0, 0);
#else
      v2f pr = { Bp[0], Bp[(size_t)N] };
      lsB2[p] = pr;
#endif
    }
    waitAsync0();              // this wave's async fills landed in LDS
    __syncthreads();           // all waves' fills visible

    if (active) {
      for (int k = 0; k < KC; k += 4) {
        v2f a0 = *(const v2f*)(Arow0 + kc + k);
        v2f a1 = *(const v2f*)(Arow1 + kc + k);
        const v2f* bp = lsB2 + ((k >> 1) + grp) * 64 + l;
        v2f b0 = bp[0], b1 = bp[16], b2 = bp[32], b3 = bp[48];
        acc[0][0] = wmma4(a0, b0, acc[0][0]);
        acc[1][0] = wmma4(a1, b0, acc[1][0]);
        acc[0][1] = wmma4(a0, b1, acc[0][1]);
        acc[1][1] = wmma4(a1, b1, acc[1][1]);
        acc[0][2] = wmma4(a0, b2, acc[0][2]);
        acc[1][2] = wmma4(a1, b2, acc[1][2]);
        acc[0][3] = wmma4(a0, b3, acc[0][3]);
        acc[1][3] = wmma4(a1, b3, acc[1][3]);
      }
    }
  }
  if (!active) return;

  // C/D layout: VGPR v, lanes 0-15 -> M=v, lanes 16-31 -> M=v+8
#pragma unroll
  for (int t = 0; t < 2; ++t) {
    float* Crow = C + (size_t)(m0 + 16 * t + 8 * grp) * N + n0 + l;
#pragma unroll
    for (int v = 0; v < 8; ++v) {
      Crow[(size_t)v * N +  0] = acc[t][0][v];
      Crow[(size_t)v * N + 16] = acc[t][1][v];
      Crow[(size_t)v * N + 32] = acc[t][2][v];
      Crow[(size_t)v * N + 48] = acc[t][3][v];
    }
  }
}

// ---------------- e[n,h] = sum_c hfeat[n, h*C+c] * att[h, c] ----------------
__global__ void attn_score(const float* __restrict__ hfeat, const float* __restrict__ att,
                           float* __restrict__ e, int N, int H, int C) {
  int idx = blockIdx.x * blockDim.x + threadIdx.x;
  if (idx >= N * H) return;
  int h = idx % H, n = idx / H;
  const float* row = hfeat + (size_t)n * H * C + (size_t)h * C;
  const float* a   = att + (size_t)h * C;
  float s = 0.f;
  for (int c = 0; c < C; ++c) s += row[c] * a[c];
  e[idx] = s;
}

__global__ void fill_f32(float* __restrict__ p, int n, float v) {
  int i = blockIdx.x * blockDim.x + threadIdx.x;
  if (i < n) p[i] = v;
}

// ---------------- edge pass 1: alpha = leaky_relu(es[src]+ed[dst]); segment max ----------------
// edges e < E come from (src[],dst[]); e >= E are self loops on node (e-E).
__global__ void edge_alpha_max(const int* __restrict__ src, const int* __restrict__ dst,
                               int E, int Nd, int H,
                               const float* __restrict__ es, const float* __restrict__ ed,
                               float* __restrict__ alpha_buf, float* __restrict__ amax) {
  int idx = blockIdx.x * blockDim.x + threadIdx.x;
  int tot = (E + Nd) * H;
  if (idx >= tot) return;
  int h = idx % H, e = idx / H;
  int s, d;
  if (e < E) { s = src[e]; d = dst[e]; } else { s = d = e - E; }
  float a = es[(size_t)s * H + h] + ed[(size_t)d * H + h];
  a = (a > 0.f) ? a : 0.2f * a;
  alpha_buf[idx] = a;
  atomMaxF(&amax[(size_t)d * H + h], a);
}

// ---------------- edge pass 2: ex = exp(alpha - amax[dst]); segment sum ----------------
__global__ void edge_exp_sum(const int* __restrict__ dst, int E, int Nd, int H,
                             float* __restrict__ alpha_buf,  // in: alpha, out: ex
                             const float* __restrict__ amax, float* __restrict__ denom) {
  int idx = blockIdx.x * blockDim.x + threadIdx.x;
  int tot = (E + Nd) * H;
  if (idx >= tot) return;
  int h = idx % H, e = idx / H;
  int d = (e < E) ? dst[e] : (e - E);
  float ex = expf(alpha_buf[idx] - amax[(size_t)d * H + h]);
  alpha_buf[idx] = ex;
  atomAddF(&denom[(size_t)d * H + h], ex);
}

// ---------------- edge pass 3: agg[dst] += (ex/denom[dst]) * hsrc[src] ----------------
// grid.x = #edges, (grid.y * blockDim.x) = H*C
__global__ void edge_aggregate(const int* __restrict__ src, const int* __restrict__ dst,
                               int E, int Nd, int H, int C,
                               const float* __restrict__ exb, const float* __restrict__ denom,
                               const float* __restrict__ hsrc, float* __restrict__ agg) {
  int e = blockIdx.x;
  int i = blockIdx.y * blockDim.x + threadIdx.x;   // 0 .. H*C-1
  int s, d;
  if (e < E) { s = src[e]; d = dst[e]; } else { s = d = e - E; }
  int h = i / C;
  float w = exb[(size_t)e * H + h] / denom[(size_t)d * H + h];
  atomAddF(&agg[(size_t)d * H * C + i], w * hsrc[(size_t)s * H * C + i]);
}

// ---------------- conv1 finalize: h1 = elu(agg + bias) ----------------
__global__ void finalize_elu(const float* __restrict__ agg, const float* __restrict__ bias,
                             float* __restrict__ out, int N, int HC) {
  int idx = blockIdx.x * blockDim.x + threadIdx.x;
  if (idx >= N * HC) return;
  float v = agg[idx] + bias[idx % HC];
  out[idx] = (v > 0.f) ? v : expm1f(v);
}

// ---------------- conv2 finalize: h2[n,c] = mean_h agg[n,h,c] + bias[c] ----------------
__global__ void finalize_mean(const float* __restrict__ agg, const float* __restrict__ bias,
                              float* __restrict__ out, int N, int H, int C) {
  int idx = blockIdx.x * blockDim.x + threadIdx.x;
  if (idx >= N * C) return;
  int n = idx / C, c = idx % C;
  float s = 0.f;
  for (int h = 0; h < H; ++h) s += agg[(size_t)n * H * C + (size_t)h * C + c];
  out[idx] = s * (1.0f / H) + bias[c];
}

// ---------------- head: out[n] = [h2 | (fg+flat_b) | last] @ out_W + out_b ----------------
__global__ void final_out(const float* __restrict__ h2, const float* __restrict__ fg,
                          const float* __restrict__ flat_b, const float* __restrict__ last,
                          const float* __restrict__ oW, const float* __restrict__ ob,
                          float* __restrict__ out, int N) {
  int n = blockIdx.x * blockDim.x + threadIdx.x;
  if (n >= N) return;
  float acc = ob[0];
  for (int c = 0; c < 64; ++c) acc += h2[(size_t)n * 64 + c] * oW[c];
  for (int c = 0; c < 64; ++c) acc += (fg[(size_t)n * 64 + c] + flat_b[c]) * oW[64 + c];
  for (int c = 0; c < 64; ++c) acc += last[(size_t)n * 64 + c] * oW[128 + c];
  out[n] = acc;
}

extern "C" void kernel_launch(void* const* d_in, const int* in_sizes, int n_in,
                              void* d_out, int out_size, void* d_ws, size_t ws_size,
                              hipStream_t stream) {
  (void)in_sizes; (void)n_in; (void)out_size; (void)ws_size;
  const float* x    = (const float*)d_in[0];
  const float* flat = (const float*)d_in[1];
  const float* last = (const float*)d_in[2];
  const int* esrc1  = (const int*)d_in[3];
  const int* edst1  = (const int*)d_in[4];
  const int* esrc2  = (const int*)d_in[5];
  const int* edst2  = (const int*)d_in[6];
  const float* W1s  = (const float*)d_in[7];
  const float* W1d  = (const float*)d_in[8];
  const float* a1s  = (const float*)d_in[9];
  const float* a1d  = (const float*)d_in[10];
  const float* b1   = (const float*)d_in[11];
  const float* W2s  = (const float*)d_in[12];
  const float* W2d  = (const float*)d_in[13];
  const float* a2s  = (const float*)d_in[14];
  const float* a2d  = (const float*)d_in[15];
  const float* b2   = (const float*)d_in[16];
  const float* fW   = (const float*)d_in[17];
  const float* fb   = (const float*)d_in[18];
  const float* oW   = (const float*)d_in[19];
  const float* ob   = (const float*)d_in[20];

  // ---- workspace layout (stage-2 scratch aliases stage-1 scratch; only h1 persists) ----
  uint8_t* ws = (uint8_t*)d_ws;
  size_t off = 0;
  auto alloc = [&](size_t bytes) -> float* {
    float* p = (float*)(ws + off);
    off += (bytes + 255) & ~(size_t)255;
    return p;
  };
  float* h1 = alloc((size_t)N2 * 512 * 4);          // persistent conv1 output
  size_t mark = off;
  // stage 1 scratch
  float* hs1   = alloc((size_t)N1 * 512 * 4);
  float* hd1   = alloc((size_t)N2 * 512 * 4);
  float* es1f  = alloc((size_t)N1 * 8 * 4);
  float* ed1f  = alloc((size_t)N2 * 8 * 4);
  float* amax1 = alloc((size_t)N2 * 8 * 4);
  float* den1  = alloc((size_t)N2 * 8 * 4);
  float* exb1  = alloc((size_t)(E1 + N2) * 8 * 4);
  float* agg1  = alloc((size_t)N2 * 512 * 4);
  // stage 2 scratch (aliases stage 1 scratch; safe: stream-ordered, stage-1 dead)
  off = mark;
  float* hs2   = alloc((size_t)N2 * 256 * 4);
  float* hd2   = alloc((size_t)N3 * 256 * 4);
  float* es2f  = alloc((size_t)N2 * 4 * 4);
  float* ed2f  = alloc((size_t)N3 * 4 * 4);
  float* amax2 = alloc((size_t)N3 * 4 * 4);
  float* den2  = alloc((size_t)N3 * 4 * 4);
  float* exb2  = alloc((size_t)(E2 + N3) * 4 * 4);
  float* agg2  = alloc((size_t)N3 * 256 * 4);
  float* h2    = alloc((size_t)N3 * 64 * 4);
  float* fg    = alloc((size_t)N3 * 64 * 4);

  const int TPB = 256;
  auto gemm_grid = [](int M, int N) { return dim3((unsigned)((M + 255) / 256), (unsigned)(N / 64)); };

  // ================= conv1 =================
  wmma_gemm_f32<<<gemm_grid(N1, 512), TPB, 0, stream>>>(x, W1s, hs1, N1, 512, 256);
  wmma_gemm_f32<<<gemm_grid(N2, 512), TPB, 0, stream>>>(x, W1d, hd1, N2, 512, 256);
  attn_score<<<(N1 * 8 + TPB - 1) / TPB, TPB, 0, stream>>>(hs1, a1s, es1f, N1, 8, 64);
  attn_score<<<(N2 * 8 + TPB - 1) / TPB, TPB, 0, stream>>>(hd1, a1d, ed1f, N2, 8, 64);
  fill_f32<<<(N2 * 8 + TPB - 1) / TPB, TPB, 0, stream>>>(amax1, N2 * 8, -1e30f);
  fill_f32<<<(N2 * 8 + TPB - 1) / TPB, TPB, 0, stream>>>(den1, N2 * 8, 0.f);
  fill_f32<<<(N2 * 512 + TPB - 1) / TPB, TPB, 0, stream>>>(agg1, N2 * 512, 0.f);
  const int ET1 = E1 + N2;  // edges + self loops
  edge_alpha_max<<<(ET1 * 8 + TPB - 1) / TPB, TPB, 0, stream>>>(esrc1, edst1, E1, N2, 8,
                                                                es1f, ed1f, exb1, amax1);
  edge_exp_sum<<<(ET1 * 8 + TPB - 1) / TPB, TPB, 0, stream>>>(edst1, E1, N2, 8,
                                                              exb1, amax1, den1);
  edge_aggregate<<<dim3(ET1, 512 / TPB), TPB, 0, stream>>>(esrc1, edst1, E1, N2, 8, 64,
                                                           exb1, den1, hs1, agg1);
  finalize_elu<<<(N2 * 512 + TPB - 1) / TPB, TPB, 0, stream>>>(agg1, b1, h1, N2, 512);

  // ================= conv2 =================
  wmma_gemm_f32<<<gemm_grid(N2, 256), TPB, 0, stream>>>(h1, W2s, hs2, N2, 256, 512);
  wmma_gemm_f32<<<gemm_grid(N3, 256), TPB, 0, stream>>>(h1, W2d, hd2, N3, 256, 512);
  attn_score<<<(N2 * 4 + TPB - 1) / TPB, TPB, 0, stream>>>(hs2, a2s, es2f, N2, 4, 64);
  attn_score<<<(N3 * 4 + TPB - 1) / TPB, TPB, 0, stream>>>(hd2, a2d, ed2f, N3, 4, 64);
  fill_f32<<<(N3 * 4 + TPB - 1) / TPB, TPB, 0, stream>>>(amax2, N3 * 4, -1e30f);
  fill_f32<<<(N3 * 4 + TPB - 1) / TPB, TPB, 0, stream>>>(den2, N3 * 4, 0.f);
  fill_f32<<<(N3 * 256 + TPB - 1) / TPB, TPB, 0, stream>>>(agg2, N3 * 256, 0.f);
  const int ET2 = E2 + N3;
  edge_alpha_max<<<(ET2 * 4 + TPB - 1) / TPB, TPB, 0, stream>>>(esrc2, edst2, E2, N3, 4,
                                                                es2f, ed2f, exb2, amax2);
  edge_exp_sum<<<(ET2 * 4 + TPB - 1) / TPB, TPB, 0, stream>>>(edst2, E2, N3, 4,
                                                              exb2, amax2, den2);
  edge_aggregate<<<dim3(ET2, 1), TPB, 0, stream>>>(esrc2, edst2, E2, N3, 4, 64,
                                                   exb2, den2, hs2, agg2);
  finalize_mean<<<(N3 * 64 + TPB - 1) / TPB, TPB, 0, stream>>>(agg2, b2, h2, N3, 4, 64);

  // ================= head =================
  wmma_gemm_f32<<<gemm_grid(N3, 64), TPB, 0, stream>>>(flat, fW, fg, N3, 64, 128);
  final_out<<<(N3 + TPB - 1) / TPB, TPB, 0, stream>>>(h2, fg, fb, last, oW, ob,
                                                      (float*)d_out, N3);
}